// GraphAggregation_69063074119736
// MI455X (gfx1250) — compile-verified
//
#include <hip/hip_runtime.h>
#include <math.h>

// ---- problem constants (from reference) ----
#define CY    64          // y channels
#define HH    80
#define WW    80
#define HD    40          // yd spatial
#define WD    40
#define M1    78          // (80-3)/1+1
#define MTOT  (78*78)     // 6084 y-patch positions
#define ND1   38          // (40-3)/1+1
#define NTOT  (38*38)     // 1444 yd-patch positions
#define KK    5
#define EPSF  1e-5f
#define OUTC  (KK*CY)     // 320
#define OUTH  160
#define OUTW  160

typedef __attribute__((ext_vector_type(2))) float v2f;
typedef __attribute__((ext_vector_type(8))) float v8f;

// -------------------------------------------------------------------------
// Kernel A: per (y-patch m, group g) mean/std over the 9 patch values.
// ystats[(m*64+g)*2 + {0,1}] = {mu, sd}
// -------------------------------------------------------------------------
__global__ void ystats_kernel(const float* __restrict__ y,
                              float* __restrict__ ystats) {
    int t = blockIdx.x * blockDim.x + threadIdx.x;
    if (t >= MTOT * CY) return;
    int mIdx = t / CY;
    int g    = t % CY;
    int i1 = mIdx / M1, i2 = mIdx % M1;
    const float* base = y + g * (HH * WW) + i1 * WW + i2;
    float s = 0.f, q = 0.f;
#pragma unroll
    for (int pi = 0; pi < 3; ++pi)
#pragma unroll
        for (int pj = 0; pj < 3; ++pj) {
            float v = base[pi * WW + pj];
            s += v; q += v * v;
        }
    float mu  = s * (1.0f / 9.0f);
    float var = (q - s * s * (1.0f / 9.0f)) * (1.0f / 8.0f); // ddof=1
    ystats[t * 2 + 0] = mu;
    ystats[t * 2 + 1] = sqrtf(var + EPSF);
}

// -------------------------------------------------------------------------
// Kernel B: per (yd-patch nIdx, group g) mean / reciprocal-std over the
// 36 values (4 sub-channels x 3x3), reduced with V_WMMA_F32_16X16X4_F32.
//   - Each wave handles 16 consecutive (nIdx,g) rows (one nIdx, 16 g's).
//   - A-matrix layout (ISA 7.12.2, 32-bit A 16x4): lanes 0-15 hold K={0,1}
//     in VGPR{0,1}; lanes 16-31 hold K={2,3}.  B = all-ones (layout-free),
//     so D rows are exact row sums; accumulate 9 steps of K=4 -> 36 elems.
//   - 92416 rows = 5776 full waves exactly -> EXEC all-ones at WMMA.
// zstats[(nIdx*64+g)*2 + {0,1}] = {mu, 1/sd}
// -------------------------------------------------------------------------
__global__ void zstats_kernel(const float* __restrict__ yd,
                              float* __restrict__ zstats) {
    int lane = threadIdx.x & 31;
    int wid  = (blockIdx.x * blockDim.x + threadIdx.x) >> 5;   // 0..5775
    int laneRow = lane & 15;
    int koff    = (lane >> 4) << 1;            // 0 for lanes 0-15, 2 for 16-31

    int nIdx = wid >> 2;                       // 16 g's per wave, 64 per nIdx
    int g    = ((wid & 3) << 4) + laneRow;
    int n1 = nIdx / ND1, n2 = nIdx % ND1;
    const float* base = yd + (4 * g) * (HD * WD) + n1 * WD + n2;

    v8f cs = {};   // sum accumulator
    v8f cq = {};   // sum-of-squares accumulator
    v2f ones; ones[0] = 1.0f; ones[1] = 1.0f;

#pragma unroll
    for (int t = 0; t < 9; ++t) {
        int k0 = 4 * t + koff;
        int k1 = k0 + 1;
        // element k -> sub-channel s = k/9, pi = (k%9)/3, pj = k%3
        int s0 = k0 / 9, r0 = k0 % 9;
        int s1 = k1 / 9, r1 = k1 % 9;
        v2f a;
        a[0] = base[s0 * (HD * WD) + (r0 / 3) * WD + (r0 % 3)];
        a[1] = base[s1 * (HD * WD) + (r1 / 3) * WD + (r1 % 3)];
        v2f a2; a2[0] = a[0] * a[0]; a2[1] = a[1] * a[1];
        cs = __builtin_amdgcn_wmma_f32_16x16x4_f32(false, a,  false, ones,
                                                   (short)0, cs, false, false);
        cq = __builtin_amdgcn_wmma_f32_16x16x4_f32(false, a2, false, ones,
                                                   (short)0, cq, false, false);
    }

    // D layout: VGPR v, lanes 0-15 -> M=v ; lanes 16-31 -> M=v+8.
    // Lane 0 writes rows 0..7, lane 16 writes rows 8..15.
    if ((lane & 15) == 0) {
        int rbase = (lane >> 4) << 3;
#pragma unroll
        for (int v = 0; v < 8; ++v) {
            int row = rbase + v;
            int Pg  = wid * 16 + row;          // == nIdx*64 + g_row
            float sum = cs[v];
            float sq  = cq[v];
            float mu  = sum * (1.0f / 36.0f);
            float var = (sq - sum * sum * (1.0f / 36.0f)) * (1.0f / 35.0f);
            zstats[Pg * 2 + 0] = mu;
            zstats[Pg * 2 + 1] = 1.0f / sqrtf(var + EPSF);
        }
    }
}

// -------------------------------------------------------------------------
// Kernel C: direct gather form of patch2im.  Output pixel (k,g,Y,X):
//   s1=Y&1, s2=X&1, qY=Y>>1, qX=X>>1 ; contributions pi,pj in [0,3) with
//   i1=qY-pi, i2=qX-pj in range.  Each contribution:
//     zd = yd[4g + (2*s1+s2), n1+pi, n2+pj],  (n1,n2)=divmod(idx[m,k],38)
//     val = (zd - z_mu)*z_rsd*y_sd + y_mu
//   out = sum / cnt.
// -------------------------------------------------------------------------
__global__ void gather_out_kernel(const float* __restrict__ yd,
                                  const int*   __restrict__ idx_k,
                                  const float* __restrict__ ystats,
                                  const float* __restrict__ zstats,
                                  float* __restrict__ out) {
    int t = blockIdx.x * blockDim.x + threadIdx.x;
    if (t >= OUTC * OUTH * OUTW) return;
    int X  = t % OUTW;
    int Y  = (t / OUTW) % OUTH;
    int oc = t / (OUTH * OUTW);
    int k  = oc / CY;
    int g  = oc % CY;

    int s1 = Y & 1, s2 = X & 1;
    int qY = Y >> 1, qX = X >> 1;
    int piLo = (qY > 77) ? (qY - 77) : 0;
    int piHi = (qY < 2) ? qY : 2;
    int pjLo = (qX > 77) ? (qX - 77) : 0;
    int pjHi = (qX < 2) ? qX : 2;

    const float* ydc = yd + (4 * g + (s1 * 2 + s2)) * (HD * WD);
    float acc = 0.0f;
    for (int pi = piLo; pi <= piHi; ++pi) {
        int i1 = qY - pi;
        for (int pj = pjLo; pj <= pjHi; ++pj) {
            int i2   = qX - pj;
            int mIdx = i1 * M1 + i2;
            int nId  = idx_k[mIdx * KK + k];
            int n1 = nId / ND1, n2 = nId % ND1;
            float zd   = ydc[(n1 + pi) * WD + (n2 + pj)];
            float zmu  = zstats[(nId * CY + g) * 2 + 0];
            float zrsd = zstats[(nId * CY + g) * 2 + 1];
            float ymu  = ystats[(mIdx * CY + g) * 2 + 0];
            float ysd  = ystats[(mIdx * CY + g) * 2 + 1];
            acc += (zd - zmu) * zrsd * ysd + ymu;
        }
    }
    float cnt = (float)((piHi - piLo + 1) * (pjHi - pjLo + 1));
    out[t] = acc / cnt;
}

// -------------------------------------------------------------------------
extern "C" void kernel_launch(void* const* d_in, const int* in_sizes, int n_in,
                              void* d_out, int out_size, void* d_ws, size_t ws_size,
                              hipStream_t stream) {
    const float* y    = (const float*)d_in[0];
    const float* yd   = (const float*)d_in[1];
    const int*   idx  = (const int*)d_in[2];
    float*       out  = (float*)d_out;

    float* ystats = (float*)d_ws;                    // 6084*64*2 floats
    float* zstats = ystats + MTOT * CY * 2;          // 1444*64*2 floats

    // A: y statistics.  389,376 threads = 1521 full blocks.
    ystats_kernel<<<(MTOT * CY + 255) / 256, 256, 0, stream>>>(y, ystats);

    // B: z statistics via f32 WMMA.  5776 waves = 722 blocks of 8 waves.
    zstats_kernel<<<722, 256, 0, stream>>>(yd, zstats);

    // C: gather-form overlap-add.  8,192,000 threads = 32,000 blocks.
    gather_out_kernel<<<(OUTC * OUTH * OUTW + 255) / 256, 256, 0, stream>>>(
        yd, idx, ystats, zstats, out);
}